// VAE_18219251270425
// MI455X (gfx1250) — compile-verified
//
#include <hip/hip_runtime.h>
#include <math.h>

// ---------------- model dimensions ----------------
#define T_LEN 200
#define BATCH 1024
#define KH    128      // hidden / embed width
#define AD    16       // action dim
#define SD    32       // state dim
#define LD    64       // latent dim
#define H3    384      // 3*K
#define ED    750

typedef float v2f __attribute__((ext_vector_type(2)));
typedef float v8f __attribute__((ext_vector_type(8)));

// Exact-precision fp32 WMMA: D(16x16) = A(16x4) x B(4x16) + C
__device__ __forceinline__ v8f wmma4(v2f a, v2f b, v8f c) {
  return __builtin_amdgcn_wmma_f32_16x16x4_f32(false, a, false, b, (short)0, c, false, false);
}

__device__ __forceinline__ float sigf(float x) { return 1.0f / (1.0f + expf(-x)); }
__device__ __forceinline__ v2f ldv2(const float* p) { return *(const v2f*)p; }

// =======================================================================
// Generic linear: Y[M,N] = act(X[M,Kd] @ W[N,Kd]^T + b)
// One wave per block; wave computes a 16-row x 128-col tile (8 n-tiles).
// Software-pipelined: fragments for step i+1 are loaded before the WMMAs
// of step i, so loads overlap the 8-WMMA burst.  Requires Kd >= 4.
// ACT: 0 = none, 1 = relu, 2 = tanh
// =======================================================================
template<int ACT>
__global__ void __launch_bounds__(32)
lin_kernel(const float* __restrict__ X, const float* __restrict__ W,
           const float* __restrict__ bias, float* __restrict__ Y,
           int Kd, int N)
{
  const int lane = threadIdx.x & 31;
  const int ml = lane & 15;
  const int hi = lane >> 4;
  const int m0 = blockIdx.x << 4;
  const int n0 = blockIdx.y << 7;

  v8f acc[8];
  #pragma unroll
  for (int j = 0; j < 8; ++j) acc[j] = (v8f){0.f,0.f,0.f,0.f,0.f,0.f,0.f,0.f};

  const float* xrow = X + (size_t)(m0 + ml) * Kd;
  const float* wp[8];
  #pragma unroll
  for (int j = 0; j < 8; ++j) {
    int n = n0 + (j << 4) + ml;
    int ns = (n < N) ? n : (N - 1);      // clamp: junk columns never stored
    wp[j] = W + (size_t)ns * Kd;
  }

  const int ks4 = Kd & ~3;
  const int koff = hi << 1;

  // software pipeline prologue
  v2f aC = ldv2(xrow + koff);
  v2f bC[8];
  #pragma unroll
  for (int j = 0; j < 8; ++j) bC[j] = ldv2(wp[j] + koff);

  for (int k = 4; k < ks4; k += 4) {
    const int kk = k + koff;
    const v2f aN = ldv2(xrow + kk);
    v2f bN[8];
    #pragma unroll
    for (int j = 0; j < 8; ++j) bN[j] = ldv2(wp[j] + kk);
    #pragma unroll
    for (int j = 0; j < 8; ++j) acc[j] = wmma4(aC, bC[j], acc[j]);
    aC = aN;
    #pragma unroll
    for (int j = 0; j < 8; ++j) bC[j] = bN[j];
  }
  #pragma unroll
  for (int j = 0; j < 8; ++j) acc[j] = wmma4(aC, bC[j], acc[j]);

  if (ks4 < Kd) {                         // wave-uniform ragged-K tail
    const float mk = hi ? 0.0f : 1.0f;    // mask reduction contribution
    v2f a = ldv2(xrow + ks4);
    a.x *= mk; a.y *= mk;
    #pragma unroll
    for (int j = 0; j < 8; ++j)
      acc[j] = wmma4(a, ldv2(wp[j] + ks4), acc[j]);
  }

  #pragma unroll
  for (int j = 0; j < 8; ++j) {
    const int n = n0 + (j << 4) + ml;
    if (n < N) {
      const float bv = bias[n];
      #pragma unroll
      for (int r = 0; r < 8; ++r) {
        const int m = m0 + r + (hi << 3);
        float v = acc[j][r] + bv;
        if (ACT == 1) v = fmaxf(v, 0.0f);
        else if (ACT == 2) v = tanhf(v);
        Y[(size_t)m * N + n] = v;
      }
    }
  }
}

// =======================================================================
// Segmented gated block (posterior): X = [X0 | X1], each segment K=128,
// total K=256, N=384.  Y = tanh(X@W1^T+b1) * sigmoid(X@W2^T+b2).
// Wave computes 16 rows x 64 cols for both weight sets (8 WMMA/k-step),
// software-pipelined within each segment.
// =======================================================================
__global__ void __launch_bounds__(32)
dgate_kernel(const float* __restrict__ X0, const float* __restrict__ X1,
             const float* __restrict__ W1, const float* __restrict__ b1,
             const float* __restrict__ W2, const float* __restrict__ b2,
             float* __restrict__ Y)
{
  const int lane = threadIdx.x & 31;
  const int ml = lane & 15;
  const int hi = lane >> 4;
  const int koff = hi << 1;
  const int m0 = blockIdx.x << 4;
  const int n0 = blockIdx.y << 6;

  v8f a1[4], a2[4];
  #pragma unroll
  for (int j = 0; j < 4; ++j) {
    a1[j] = (v8f){0.f,0.f,0.f,0.f,0.f,0.f,0.f,0.f};
    a2[j] = a1[j];
  }
  const float* wa[4];
  const float* wb[4];
  #pragma unroll
  for (int j = 0; j < 4; ++j) {
    const int n = n0 + (j << 4) + ml;
    wa[j] = W1 + (size_t)n * 256;
    wb[j] = W2 + (size_t)n * 256;
  }
  const float* xs[2] = {X0, X1};
  #pragma unroll
  for (int s = 0; s < 2; ++s) {
    const float* xrow = xs[s] + (size_t)(m0 + ml) * KH;
    const int kb = s << 7;
    // pipeline prologue
    v2f aC = ldv2(xrow + koff);
    v2f bCa[4], bCb[4];
    #pragma unroll
    for (int j = 0; j < 4; ++j) {
      bCa[j] = ldv2(wa[j] + kb + koff);
      bCb[j] = ldv2(wb[j] + kb + koff);
    }
    for (int k = 4; k < 128; k += 4) {
      const int kk = k + koff;
      const v2f aN = ldv2(xrow + kk);
      v2f bNa[4], bNb[4];
      #pragma unroll
      for (int j = 0; j < 4; ++j) {
        bNa[j] = ldv2(wa[j] + kb + kk);
        bNb[j] = ldv2(wb[j] + kb + kk);
      }
      #pragma unroll
      for (int j = 0; j < 4; ++j) {
        a1[j] = wmma4(aC, bCa[j], a1[j]);
        a2[j] = wmma4(aC, bCb[j], a2[j]);
      }
      aC = aN;
      #pragma unroll
      for (int j = 0; j < 4; ++j) { bCa[j] = bNa[j]; bCb[j] = bNb[j]; }
    }
    #pragma unroll
    for (int j = 0; j < 4; ++j) {
      a1[j] = wmma4(aC, bCa[j], a1[j]);
      a2[j] = wmma4(aC, bCb[j], a2[j]);
    }
  }
  #pragma unroll
  for (int j = 0; j < 4; ++j) {
    const int n = n0 + (j << 4) + ml;
    const float bv1 = b1[n];
    const float bv2 = b2[n];
    #pragma unroll
    for (int r = 0; r < 8; ++r) {
      const int m = m0 + r + (hi << 3);
      Y[(size_t)m * H3 + n] = tanhf(a1[j][r] + bv1) * sigf(a2[j][r] + bv2);
    }
  }
}

// =======================================================================
// Dual head (N=64) with fused reparameterization sampling.
// MODE 0 (posterior): Y1=mu, Y2=ls, Y3 = mu + exp(0.5*ls)*EPS
// MODE 1 (encoder):   Y1=mean, Y2=std=exp(clip(ls,-4,15)), Y3=mean+std*EPS
// Wave computes all 64 cols for both weight sets; software-pipelined.
// =======================================================================
template<int MODE>
__global__ void __launch_bounds__(32)
dhead_kernel(const float* __restrict__ X,
             const float* __restrict__ W1, const float* __restrict__ b1,
             const float* __restrict__ W2, const float* __restrict__ b2,
             float* __restrict__ Y1, float* __restrict__ Y2,
             float* __restrict__ Y3, const float* __restrict__ EPS, int Kd)
{
  const int lane = threadIdx.x & 31;
  const int ml = lane & 15;
  const int hi = lane >> 4;
  const int koff = hi << 1;
  const int m0 = blockIdx.x << 4;

  v8f acc1[4], acc2[4];
  #pragma unroll
  for (int j = 0; j < 4; ++j) {
    acc1[j] = (v8f){0.f,0.f,0.f,0.f,0.f,0.f,0.f,0.f};
    acc2[j] = acc1[j];
  }
  const float* xrow = X + (size_t)(m0 + ml) * Kd;
  const float* wa[4];
  const float* wb[4];
  #pragma unroll
  for (int j = 0; j < 4; ++j) {
    const int n = (j << 4) + ml;
    wa[j] = W1 + (size_t)n * Kd;
    wb[j] = W2 + (size_t)n * Kd;
  }
  const int ks4 = Kd & ~3;

  v2f aC = ldv2(xrow + koff);
  v2f bCa[4], bCb[4];
  #pragma unroll
  for (int j = 0; j < 4; ++j) {
    bCa[j] = ldv2(wa[j] + koff);
    bCb[j] = ldv2(wb[j] + koff);
  }
  for (int k = 4; k < ks4; k += 4) {
    const int kk = k + koff;
    const v2f aN = ldv2(xrow + kk);
    v2f bNa[4], bNb[4];
    #pragma unroll
    for (int j = 0; j < 4; ++j) {
      bNa[j] = ldv2(wa[j] + kk);
      bNb[j] = ldv2(wb[j] + kk);
    }
    #pragma unroll
    for (int j = 0; j < 4; ++j) {
      acc1[j] = wmma4(aC, bCa[j], acc1[j]);
      acc2[j] = wmma4(aC, bCb[j], acc2[j]);
    }
    aC = aN;
    #pragma unroll
    for (int j = 0; j < 4; ++j) { bCa[j] = bNa[j]; bCb[j] = bNb[j]; }
  }
  #pragma unroll
  for (int j = 0; j < 4; ++j) {
    acc1[j] = wmma4(aC, bCa[j], acc1[j]);
    acc2[j] = wmma4(aC, bCb[j], acc2[j]);
  }
  if (ks4 < Kd) {                       // ragged K tail (Kd=750)
    const float mk = hi ? 0.0f : 1.0f;
    v2f a = ldv2(xrow + ks4);
    a.x *= mk; a.y *= mk;
    #pragma unroll
    for (int j = 0; j < 4; ++j) {
      acc1[j] = wmma4(a, ldv2(wa[j] + ks4), acc1[j]);
      acc2[j] = wmma4(a, ldv2(wb[j] + ks4), acc2[j]);
    }
  }

  #pragma unroll
  for (int j = 0; j < 4; ++j) {
    const int n = (j << 4) + ml;
    const float bv1 = b1[n];
    const float bv2 = b2[n];
    #pragma unroll
    for (int r = 0; r < 8; ++r) {
      const int m = m0 + r + (hi << 3);
      const size_t idx = (size_t)m * LD + n;
      const float v1 = acc1[j][r] + bv1;
      const float v2 = acc2[j][r] + bv2;
      if (MODE == 0) {
        Y1[idx] = v1;
        Y2[idx] = v2;
        Y3[idx] = v1 + expf(0.5f * v2) * EPS[idx];
      } else {
        const float lc = fminf(fmaxf(v2, -4.0f), 15.0f);
        const float s = expf(lc);
        Y1[idx] = v1;
        Y2[idx] = s;
        Y3[idx] = v1 + s * EPS[idx];
      }
    }
  }
}

// =======================================================================
// Fully fused GRU cell:
//   gi = [xe|ue|ze] @ Wih^T + bih   (K=3x128 segments, N=384)
//   gh = h @ Whh^T + bhh            (K=128, N=384)
//   h' = (1-z)*tanh(i_n + r*h_n) + z*h,  r/z from sigmoid
// Wave computes a 16-row x 32-col tile of h'; holds r/z/n gate tiles for
// both gi and gh (12 accumulators); software-pipelined per segment.
// Writes h' to h_seq AND rnn_seq.
// =======================================================================
__global__ void __launch_bounds__(32)
grucell_kernel(const float* __restrict__ xe, const float* __restrict__ ue,
               const float* __restrict__ ze, const float* __restrict__ h,
               const float* __restrict__ Wih, const float* __restrict__ bih,
               const float* __restrict__ Whh, const float* __restrict__ bhh,
               float* __restrict__ hn, float* __restrict__ hn2)
{
  const int lane = threadIdx.x & 31;
  const int ml = lane & 15;
  const int hi = lane >> 4;
  const int koff = hi << 1;
  const int m0 = blockIdx.x << 4;
  const int c0 = blockIdx.y << 5;

  v8f gia[3][2], gha[3][2];
  #pragma unroll
  for (int g = 0; g < 3; ++g)
    #pragma unroll
    for (int j = 0; j < 2; ++j) {
      gia[g][j] = (v8f){0.f,0.f,0.f,0.f,0.f,0.f,0.f,0.f};
      gha[g][j] = gia[g][j];
    }

  const float* wi[3][2];
  #pragma unroll
  for (int g = 0; g < 3; ++g)
    #pragma unroll
    for (int j = 0; j < 2; ++j) {
      const int n = (g << 7) + c0 + (j << 4) + ml;
      wi[g][j] = Wih + (size_t)n * H3;
    }

  // gi over the three concat segments (no materialized concat)
  const float* segs[3] = {xe, ue, ze};
  #pragma unroll
  for (int s = 0; s < 3; ++s) {
    const float* xrow = segs[s] + (size_t)(m0 + ml) * KH;
    const int kb = s << 7;
    v2f aC = ldv2(xrow + koff);
    v2f bC[3][2];
    #pragma unroll
    for (int g = 0; g < 3; ++g)
      #pragma unroll
      for (int j = 0; j < 2; ++j) bC[g][j] = ldv2(wi[g][j] + kb + koff);
    for (int k = 4; k < 128; k += 4) {
      const int kk = k + koff;
      const v2f aN = ldv2(xrow + kk);
      v2f bN[3][2];
      #pragma unroll
      for (int g = 0; g < 3; ++g)
        #pragma unroll
        for (int j = 0; j < 2; ++j) bN[g][j] = ldv2(wi[g][j] + kb + kk);
      #pragma unroll
      for (int g = 0; g < 3; ++g)
        #pragma unroll
        for (int j = 0; j < 2; ++j) gia[g][j] = wmma4(aC, bC[g][j], gia[g][j]);
      aC = aN;
      #pragma unroll
      for (int g = 0; g < 3; ++g)
        #pragma unroll
        for (int j = 0; j < 2; ++j) bC[g][j] = bN[g][j];
    }
    #pragma unroll
    for (int g = 0; g < 3; ++g)
      #pragma unroll
      for (int j = 0; j < 2; ++j) gia[g][j] = wmma4(aC, bC[g][j], gia[g][j]);
  }

  // gh over h
  {
    const float* wh[3][2];
    #pragma unroll
    for (int g = 0; g < 3; ++g)
      #pragma unroll
      for (int j = 0; j < 2; ++j) {
        const int n = (g << 7) + c0 + (j << 4) + ml;
        wh[g][j] = Whh + (size_t)n * KH;
      }
    const float* hrow = h + (size_t)(m0 + ml) * KH;
    v2f aC = ldv2(hrow + koff);
    v2f bC[3][2];
    #pragma unroll
    for (int g = 0; g < 3; ++g)
      #pragma unroll
      for (int j = 0; j < 2; ++j) bC[g][j] = ldv2(wh[g][j] + koff);
    for (int k = 4; k < 128; k += 4) {
      const int kk = k + koff;
      const v2f aN = ldv2(hrow + kk);
      v2f bN[3][2];
      #pragma unroll
      for (int g = 0; g < 3; ++g)
        #pragma unroll
        for (int j = 0; j < 2; ++j) bN[g][j] = ldv2(wh[g][j] + kk);
      #pragma unroll
      for (int g = 0; g < 3; ++g)
        #pragma unroll
        for (int j = 0; j < 2; ++j) gha[g][j] = wmma4(aC, bC[g][j], gha[g][j]);
      aC = aN;
      #pragma unroll
      for (int g = 0; g < 3; ++g)
        #pragma unroll
        for (int j = 0; j < 2; ++j) bC[g][j] = bN[g][j];
    }
    #pragma unroll
    for (int g = 0; g < 3; ++g)
      #pragma unroll
      for (int j = 0; j < 2; ++j) gha[g][j] = wmma4(aC, bC[g][j], gha[g][j]);
  }

  // GRU nonlinearity + dual store
  #pragma unroll
  for (int j = 0; j < 2; ++j) {
    const int c = c0 + (j << 4) + ml;
    const float bir = bih[c],       bhr = bhh[c];
    const float biz = bih[128 + c], bhz = bhh[128 + c];
    const float bin = bih[256 + c], bhn = bhh[256 + c];
    #pragma unroll
    for (int r = 0; r < 8; ++r) {
      const int m = m0 + r + (hi << 3);
      const float rr = sigf((gia[0][j][r] + bir) + (gha[0][j][r] + bhr));
      const float zz = sigf((gia[1][j][r] + biz) + (gha[1][j][r] + bhz));
      const float nn = tanhf((gia[2][j][r] + bin) + rr * (gha[2][j][r] + bhn));
      const float hv = h[(size_t)m * KH + c];
      const float o = (1.0f - zz) * nn + zz * hv;
      hn[(size_t)m * KH + c] = o;
      hn2[(size_t)m * KH + c] = o;
    }
  }
}

// =======================================================================
// Fused 2-layer MLP (relu(relu(X@W1^T+b1)@W2^T+b2)), N1=N2=128, K1 is a
// template constant (16/32/64) so the layer-1 K loop fully unrolls (layer 2 by 8), guard-free.
// 4 waves per block, 16 rows per block; layer-1 tile staged in LDS.
// =======================================================================
template<int K1>
__global__ void __launch_bounds__(128)
mlp2_kernel(const float* __restrict__ X,
            const float* __restrict__ W1, const float* __restrict__ b1,
            const float* __restrict__ W2, const float* __restrict__ b2,
            float* __restrict__ Y)
{
  __shared__ float h1[16 * 130];
  const int lane = threadIdx.x & 31;
  const int wave = threadIdx.x >> 5;
  const int ml = lane & 15;
  const int hi = lane >> 4;
  const int m0 = blockIdx.x << 4;
  const int koff = hi << 1;

  // ---- layer 1: this wave computes cols [32*wave, 32*wave+32) ----
  v8f a0 = (v8f){0.f,0.f,0.f,0.f,0.f,0.f,0.f,0.f};
  v8f a1 = a0;
  const float* xrow = X + (size_t)(m0 + ml) * K1;
  const int nA = (wave << 5) + ml;
  const float* w0 = W1 + (size_t)nA * K1;
  const float* w1 = W1 + (size_t)(nA + 16) * K1;
  #pragma unroll
  for (int k = 0; k < K1; k += 4) {
    const int kk = k + koff;
    const v2f a = ldv2(xrow + kk);
    a0 = wmma4(a, ldv2(w0 + kk), a0);
    a1 = wmma4(a, ldv2(w1 + kk), a1);
  }
  {
    const float bv0 = b1[nA];
    const float bv1 = b1[nA + 16];
    #pragma unroll
    for (int r = 0; r < 8; ++r) {
      const int mrow = r + (hi << 3);
      h1[mrow * 130 + nA]      = fmaxf(a0[r] + bv0, 0.0f);
      h1[mrow * 130 + nA + 16] = fmaxf(a1[r] + bv1, 0.0f);
    }
  }
  __syncthreads();

  // ---- layer 2: K=128 from LDS ----
  v8f c0 = (v8f){0.f,0.f,0.f,0.f,0.f,0.f,0.f,0.f};
  v8f c1 = c0;
  const float* w2a = W2 + (size_t)nA * 128;
  const float* w2b = W2 + (size_t)(nA + 16) * 128;
  #pragma unroll 8
  for (int k = 0; k < 128; k += 4) {
    const int kk = k + koff;
    v2f a;
    a.x = h1[ml * 130 + kk];
    a.y = h1[ml * 130 + kk + 1];
    c0 = wmma4(a, ldv2(w2a + kk), c0);
    c1 = wmma4(a, ldv2(w2b + kk), c1);
  }
  {
    const float bv0 = b2[nA];
    const float bv1 = b2[nA + 16];
    #pragma unroll
    for (int r = 0; r < 8; ++r) {
      const int m = m0 + r + (hi << 3);
      Y[(size_t)m * 128 + nA]      = fmaxf(c0[r] + bv0, 0.0f);
      Y[(size_t)m * 128 + nA + 16] = fmaxf(c1[r] + bv1, 0.0f);
    }
  }
}

// ---------------- elementwise helpers ----------------
__global__ void fill_kernel(float* p, int n, float v) {
  int i = blockIdx.x * 256 + threadIdx.x;
  if (i < n) p[i] = v;
}
__global__ void copyseg_kernel(float* dst, int dstStride, int dstOff,
                               const float* src, int srcStride, int n, int cols) {
  int i = blockIdx.x * 256 + threadIdx.x;
  if (i < n) {
    int r = i / cols, c = i - r * cols;
    dst[(size_t)r * dstStride + dstOff + c] = src[(size_t)r * srcStride + c];
  }
}

// ---------------- host-side launch helpers ----------------
static inline int eg(int n) { return (n + 255) / 256; }

static void lin_go(int act, const float* X, const float* W, const float* b, float* Y,
                   int M, int Kd, int N, hipStream_t s) {
  dim3 g(M / 16, (N + 127) / 128), blk(32);
  if (act == 0)      lin_kernel<0><<<g, blk, 0, s>>>(X, W, b, Y, Kd, N);
  else if (act == 1) lin_kernel<1><<<g, blk, 0, s>>>(X, W, b, Y, Kd, N);
  else               lin_kernel<2><<<g, blk, 0, s>>>(X, W, b, Y, Kd, N);
}

static void mlp2_go(const float* X, const float* W1, const float* b1,
                    const float* W2, const float* b2, float* Y,
                    int M, int K1, hipStream_t s) {
  dim3 g(M / 16), blk(128);
  if (K1 == 16)      mlp2_kernel<16><<<g, blk, 0, s>>>(X, W1, b1, W2, b2, Y);
  else if (K1 == 32) mlp2_kernel<32><<<g, blk, 0, s>>>(X, W1, b1, W2, b2, Y);
  else               mlp2_kernel<64><<<g, blk, 0, s>>>(X, W1, b1, W2, b2, Y);
}

extern "C" void kernel_launch(void* const* d_in, const int* in_sizes, int n_in,
                              void* d_out, int out_size, void* d_ws, size_t ws_size,
                              hipStream_t stream) {
  const float* in[46];
  for (int i = 0; i < n_in && i < 46; ++i) in[i] = (const float*)d_in[i];

  const float *ext, *obs, *epz, *epe;
  const float *puW1,*pub1,*puW2,*pub2, *pxW1,*pxb1,*pxW2,*pxb2;
  const float *pzW1,*pzb1,*pzW2,*pzb2, *peW1,*peb1,*peW2,*peb2;
  const float *poW1,*pob1,*poW2,*pob2, *Wmu,*bmu,*Wls,*bls;
  const float *Wih,*bih,*Whh,*bhh;
  const float *e1W,*e1b,*e2W,*e2b,*meW,*meb,*lsW,*lsb;
  const float *d1W,*d1b,*d2W,*d2b,*d3W,*d3b;

  if (in_sizes[0] == T_LEN * BATCH * AD) {
    // dict insertion order
    ext=in[0]; obs=in[1]; epz=in[2]; epe=in[3];
    puW1=in[4];  pub1=in[5];  puW2=in[6];  pub2=in[7];
    pxW1=in[8];  pxb1=in[9];  pxW2=in[10]; pxb2=in[11];
    pzW1=in[12]; pzb1=in[13]; pzW2=in[14]; pzb2=in[15];
    peW1=in[16]; peb1=in[17]; peW2=in[18]; peb2=in[19];
    poW1=in[20]; pob1=in[21]; poW2=in[22]; pob2=in[23];
    Wmu=in[24];  bmu=in[25];  Wls=in[26];  bls=in[27];
    Wih=in[28];  bih=in[29];  Whh=in[30];  bhh=in[31];
    e1W=in[32];  e1b=in[33];  e2W=in[34];  e2b=in[35];
    meW=in[36];  meb=in[37];  lsW=in[38];  lsb=in[39];
    d1W=in[40];  d1b=in[41];  d2W=in[42];  d2b=in[43];
    d3W=in[44];  d3b=in[45];
  } else {
    // jax pytree (sorted-key) order
    epe=in[0]; epz=in[1]; ext=in[2]; obs=in[3];
    Whh=in[4]; Wih=in[5]; bhh=in[6]; bih=in[7];
    d1W=in[8];  d1b=in[9];  d2W=in[10]; d2b=in[11]; d3W=in[12]; d3b=in[13];
    e1W=in[14]; e1b=in[15]; e2W=in[16]; e2b=in[17];
    lsW=in[18]; lsb=in[19]; meW=in[20]; meb=in[21];
    peW1=in[22]; peW2=in[23]; peb1=in[24]; peb2=in[25];
    poW1=in[26]; poW2=in[27]; Wls=in[28]; Wmu=in[29];
    pob1=in[30]; pob2=in[31]; bls=in[32]; bmu=in[33];
    puW1=in[34]; puW2=in[35]; pub1=in[36]; pub2=in[37];
    pxW1=in[38]; pxW2=in[39]; pxb1=in[40]; pxb2=in[41];
    pzW1=in[42]; pzW2=in[43]; pzb1=in[44]; pzb2=in[45];
  }

  float* out = (float*)d_out;
  // output region offsets (floats), in reference return order
  const size_t O_at  = 0;
  const size_t O_et  = O_at  + (size_t)BATCH * AD;
  const size_t O_em  = O_et  + (size_t)BATCH * LD;
  const size_t O_es  = O_em  + (size_t)BATCH * LD;
  const size_t O_mu  = O_es  + (size_t)BATCH * LD;
  const size_t O_ls  = O_mu  + (size_t)T_LEN * BATCH * LD;
  const size_t O_z   = O_ls  + (size_t)T_LEN * BATCH * LD;
  const size_t O_h   = O_z   + (size_t)T_LEN * BATCH * LD;
  const size_t O_u   = O_h   + (size_t)(T_LEN + 1) * BATCH * KH;
  const size_t O_rnn = O_u   + (size_t)T_LEN * BATCH * KH;

  // workspace layout (floats)
  float* ws = (float*)d_ws;
  size_t o = 0;
  float* xEmb = ws + o; o += (size_t)T_LEN * BATCH * KH;
  float* Tt   = ws + o; o += (size_t)BATCH * H3;
  float* ze   = ws + o; o += (size_t)BATCH * KH;
  float* ztE  = ws + o; o += (size_t)BATCH * KH;
  float* feat = ws + o; o += (size_t)BATCH * 512;
  float* eb1  = ws + o; o += (size_t)BATCH * 768;
  float* eb2  = ws + o; o += (size_t)BATCH * 768;
  float* eEmb = ws + o; o += (size_t)BATCH * KH;
  float* ain  = ws + o; o += (size_t)BATCH * 512;
  float* ab1  = ws + o; o += (size_t)BATCH * 768;
  float* ab2  = ws + o; o += (size_t)BATCH * 768;

  const int MB = BATCH;

  // ---- preprocess over full sequence ----
  mlp2_go(ext, puW1, pub1, puW2, pub2, out + O_u, T_LEN * BATCH, AD, stream);
  mlp2_go(obs, pxW1, pxb1, pxW2, pxb2, xEmb,      T_LEN * BATCH, SD, stream);
  // h0 = 0 in both h_seq and rnn_hidden_state_seq
  fill_kernel<<<eg(MB * KH), 256, 0, stream>>>(out + O_h,   MB * KH, 0.0f);
  fill_kernel<<<eg(MB * KH), 256, 0, stream>>>(out + O_rnn, MB * KH, 0.0f);

  // ---- sequential scan: 4 fused kernels per step ----
  for (int t = 0; t < T_LEN; ++t) {
    const float* xe = xEmb + (size_t)t * MB * KH;
    const float* ue = out + O_u + (size_t)t * MB * KH;
    const float* h  = out + O_h + (size_t)t * MB * KH;
    float* hn       = out + O_h   + (size_t)(t + 1) * MB * KH;
    float* hn2      = out + O_rnn + (size_t)(t + 1) * MB * KH;
    float* muT = out + O_mu + (size_t)t * MB * LD;
    float* lsT = out + O_ls + (size_t)t * MB * LD;
    float* zT  = out + O_z  + (size_t)t * MB * LD;

    // gated posterior block over [xe|h]
    dgate_kernel<<<dim3(MB / 16, H3 / 64), 32, 0, stream>>>(
        xe, h, poW1, pob1, poW2, pob2, Tt);
    // mu / log-sigma heads + z-sample fused
    dhead_kernel<0><<<dim3(MB / 16), 32, 0, stream>>>(
        Tt, Wmu, bmu, Wls, bls, muT, lsT, zT,
        epz + (size_t)t * MB * LD, H3);
    // ze = preprocess_z(z)
    mlp2_go(zT, pzW1, pzb1, pzW2, pzb2, ze, MB, LD, stream);
    // fused GRU cell over [xe|ue|ze] and h; writes h_seq and rnn_seq
    grucell_kernel<<<dim3(MB / 16, KH / 32), 32, 0, stream>>>(
        xe, ue, ze, h, Wih, bih, Whh, bhh, hn, hn2);
  }

  // ---- tail ----
  const float* xeT = xEmb + (size_t)(T_LEN - 1) * MB * KH;
  const float* ueT = out + O_u + (size_t)(T_LEN - 1) * MB * KH;
  const float* hT  = out + O_h + (size_t)(T_LEN - 1) * MB * KH;   // h_seq[-2]
  const float* zT  = out + O_z + (size_t)(T_LEN - 1) * MB * LD;

  mlp2_go(zT, pzW1, pzb1, pzW2, pzb2, ztE, MB, LD, stream);

  copyseg_kernel<<<eg(MB * KH), 256, 0, stream>>>(feat, 512, 0,   hT,  KH, MB * KH, KH);
  copyseg_kernel<<<eg(MB * KH), 256, 0, stream>>>(feat, 512, 128, xeT, KH, MB * KH, KH);
  copyseg_kernel<<<eg(MB * KH), 256, 0, stream>>>(feat, 512, 256, ueT, KH, MB * KH, KH);
  copyseg_kernel<<<eg(MB * KH), 256, 0, stream>>>(feat, 512, 384, ztE, KH, MB * KH, KH);

  lin_go(1, feat, e1W, e1b, eb1, MB, 512, ED, stream);
  lin_go(1, eb1,  e2W, e2b, eb2, MB, ED,  ED, stream);
  // mean / log_std heads + clip/exp/e-sample fused
  dhead_kernel<1><<<dim3(MB / 16), 32, 0, stream>>>(
      eb2, meW, meb, lsW, lsb, out + O_em, out + O_es, out + O_et, epe, ED);

  mlp2_go(out + O_et, peW1, peb1, peW2, peb2, eEmb, MB, LD, stream);

  copyseg_kernel<<<eg(MB * KH), 256, 0, stream>>>(ain, 512, 0,   hT,   KH, MB * KH, KH);
  copyseg_kernel<<<eg(MB * KH), 256, 0, stream>>>(ain, 512, 128, eEmb, KH, MB * KH, KH);
  copyseg_kernel<<<eg(MB * KH), 256, 0, stream>>>(ain, 512, 256, xeT,  KH, MB * KH, KH);
  copyseg_kernel<<<eg(MB * KH), 256, 0, stream>>>(ain, 512, 384, ztE,  KH, MB * KH, KH);

  lin_go(1, ain, d1W, d1b, ab1, MB, 512, ED, stream);
  lin_go(1, ab1, d2W, d2b, ab2, MB, ED,  ED, stream);
  lin_go(2, ab2, d3W, d3b, out + O_at, MB, ED, AD, stream);   // a_t = 1.0 * tanh(...)

  (void)out_size; (void)ws_size; (void)n_in;
}